// GNNLayer_18511309046193
// MI455X (gfx1250) — compile-verified
//
#include <hip/hip_runtime.h>

#define B_ 4
#define V_ 256
#define H_ 128
#define EPS_ 1e-5f

typedef __attribute__((ext_vector_type(2))) float v2f;
typedef __attribute__((ext_vector_type(8))) float v8f;

// ---------------------------------------------------------------------------
// Kernel 1: Uh, Vn, Ah, Bh = h @ {U,V,A,B}w.T + bias    (exact fp32, tiny)
// grid = B*V blocks, 128 threads
// ---------------------------------------------------------------------------
__global__ void __launch_bounds__(128)
pre_linears(const float* __restrict__ h,
            const float* __restrict__ Uw, const float* __restrict__ Ub,
            const float* __restrict__ Vw, const float* __restrict__ Vb,
            const float* __restrict__ Aw, const float* __restrict__ Ab,
            const float* __restrict__ Bw, const float* __restrict__ Bb,
            float* __restrict__ Uh, float* __restrict__ Vn,
            float* __restrict__ Ah, float* __restrict__ Bh)
{
    __shared__ float hrow[H_];
    const int row = blockIdx.x;
    const int t   = threadIdx.x;
    hrow[t] = h[row * H_ + t];
    __syncthreads();

    float su = Ub[t], sv = Vb[t], sa = Ab[t], sb = Bb[t];
    const float* uw = Uw + t * H_;
    const float* vw = Vw + t * H_;
    const float* aw = Aw + t * H_;
    const float* bw = Bw + t * H_;
#pragma unroll 8
    for (int k = 0; k < H_; ++k) {
        float x = hrow[k];
        su = fmaf(x, uw[k], su);
        sv = fmaf(x, vw[k], sv);
        sa = fmaf(x, aw[k], sa);
        sb = fmaf(x, bw[k], sb);
    }
    Uh[row * H_ + t] = su;
    Vn[row * H_ + t] = sv;
    Ah[row * H_ + t] = sa;
    Bh[row * H_ + t] = sb;
}

// ---------------------------------------------------------------------------
// Kernel 2: fused edge path, exact fp32 via V_WMMA_F32_16X16X4_F32.
// One workgroup per (b,i); 8 waves, wave w owns h-channels [16w,16w+16).
// Per wave: B fragments (Cw row ch, 32 K-chunks x v2f) preloaded in registers;
// per 64-row j-block: LDS-staged e tile (+prefetch of the next tile),
// 4 output tiles computed concurrently (4 independent WMMA chains);
// fused epilogue: e_new = Ce + Cb + Ah[j] + Bh[i] -> d_out, sigmoid-gated agg,
// per-channel BN partial sums.
// ---------------------------------------------------------------------------
__global__ void __launch_bounds__(256)
fused_edge(const float* __restrict__ e,
           const float* __restrict__ Cw, const float* __restrict__ Cb,
           const float* __restrict__ Ah, const float* __restrict__ Bh,
           const float* __restrict__ Vn,
           float* __restrict__ enew,     // d_out e-region (B,V,V,H)
           float* __restrict__ agg,      // (B,V,H)
           float* __restrict__ part_e)   // (B*V, 256) [sum | sumsq]
{
    const int wg   = blockIdx.x;        // 0..1023
    const int b    = wg / V_;
    const int i    = wg % V_;
    const int t    = threadIdx.x;
    const int w    = t >> 5;            // wave id 0..7
    const int lane = t & 31;
    const int half = lane >> 4;         // 0: lanes 0-15, 1: lanes 16-31
    const int lq   = lane & 15;
    const int ch   = w * 16 + lq;       // this lane's h channel (C/D: N = lane%16)

    __shared__ float tile[64][132];     // 64 j-rows x 128 k, padded (bank-conflict-free)

    // ---- B fragments, preloaded once: B[k][n] = Cw[n][k], n = ch.
    // f32 4x16 B layout: VGPR0: K=0 (lanes 0-15) / K=2 (lanes 16-31); VGPR1: K=1 / K=3
    // -> lane loads contiguous pair Cw[ch][4*kk + 2*half + {0,1}]
    v2f bfr[32];
#pragma unroll
    for (int kk = 0; kk < 32; ++kk) {
        bfr[kk] = *(const v2f*)(Cw + ch * H_ + 4 * kk + 2 * half);
    }

    const float bhv = Bh[(b * V_ + i) * H_ + ch] + Cb[ch];
    float aggv = 0.f, ssum = 0.f, ssq = 0.f;

    const float* ebase = e    + (size_t)(b * V_ + i) * V_ * H_;
    float*       obase = enew + (size_t)(b * V_ + i) * V_ * H_;

    for (int jb = 0; jb < 4; ++jb) {
        __syncthreads();
        // stage e[b,i, jb*64 .. jb*64+63, :] into LDS (coalesced float4)
        const float4* src = (const float4*)(ebase + (size_t)jb * 64 * H_);
        for (int idx = t; idx < 64 * 32; idx += 256) {
            const int row = idx >> 5, c4 = idx & 31;
            *(float4*)&tile[row][c4 * 4] = src[row * 32 + c4];
        }
        // prefetch next j-block's tile into cache while we compute this one
        if (jb < 3) {
            const float* nxt = ebase + (size_t)(jb + 1) * 64 * H_;
            __builtin_prefetch(nxt + (size_t)t * 32, 0, 3);   // 256 thr x 128B = 32KB
        }
        __syncthreads();

        // 4 concurrent 16x16 output tiles (jt = 0..3): independent WMMA chains
        v8f acc0 = {0.f,0.f,0.f,0.f,0.f,0.f,0.f,0.f};
        v8f acc1 = {0.f,0.f,0.f,0.f,0.f,0.f,0.f,0.f};
        v8f acc2 = {0.f,0.f,0.f,0.f,0.f,0.f,0.f,0.f};
        v8f acc3 = {0.f,0.f,0.f,0.f,0.f,0.f,0.f,0.f};
        // A 16x4 f32 layout: M = lane%16; lane pair K = 2*half + {0,1}
        const float* tr0 = &tile[ 0 + lq][2 * half];
        const float* tr1 = &tile[16 + lq][2 * half];
        const float* tr2 = &tile[32 + lq][2 * half];
        const float* tr3 = &tile[48 + lq][2 * half];
#pragma unroll
        for (int kk = 0; kk < 32; ++kk) {
            const v2f a0 = *(const v2f*)(tr0 + 4 * kk);
            const v2f a1 = *(const v2f*)(tr1 + 4 * kk);
            const v2f a2 = *(const v2f*)(tr2 + 4 * kk);
            const v2f a3 = *(const v2f*)(tr3 + 4 * kk);
            acc0 = __builtin_amdgcn_wmma_f32_16x16x4_f32(false, a0, false, bfr[kk],
                                                         (short)0, acc0, false, false);
            acc1 = __builtin_amdgcn_wmma_f32_16x16x4_f32(false, a1, false, bfr[kk],
                                                         (short)0, acc1, false, false);
            acc2 = __builtin_amdgcn_wmma_f32_16x16x4_f32(false, a2, false, bfr[kk],
                                                         (short)0, acc2, false, false);
            acc3 = __builtin_amdgcn_wmma_f32_16x16x4_f32(false, a3, false, bfr[kk],
                                                         (short)0, acc3, false, false);
        }

        // epilogue: C/D layout -> VGPR r: M = r + 8*half
#pragma unroll
        for (int jt = 0; jt < 4; ++jt) {
            const v8f acc = (jt == 0) ? acc0 : (jt == 1) ? acc1 : (jt == 2) ? acc2 : acc3;
            const int jrow0 = jb * 64 + jt * 16 + (half ? 8 : 0);
#pragma unroll
            for (int r = 0; r < 8; ++r) {
                const int j = jrow0 + r;
                const float en = acc[r] + bhv + Ah[(b * V_ + j) * H_ + ch];
                obase[(size_t)j * H_ + ch] = en;
                const float g = 1.f / (1.f + __expf(-en));
                aggv = fmaf(g, Vn[(b * V_ + j) * H_ + ch], aggv);
                ssum += en;
                ssq  = fmaf(en, en, ssq);
            }
        }
    }

    // lanes L and L^16 hold the same channel, disjoint j sets -> combine halves
    aggv += __shfl_xor(aggv, 16, 32);
    ssum += __shfl_xor(ssum, 16, 32);
    ssq  += __shfl_xor(ssq,  16, 32);
    if (half == 0) {
        agg[(b * V_ + i) * H_ + ch] = aggv;
        part_e[wg * 256 + ch]       = ssum;
        part_e[wg * 256 + 128 + ch] = ssq;
    }
}

// ---------------------------------------------------------------------------
// Kernel 3: fold 1024 WG partials -> per-channel BN scale/shift for e
// ---------------------------------------------------------------------------
__global__ void __launch_bounds__(128)
stats_e(const float* __restrict__ part,
        const float* __restrict__ gamma_e, const float* __restrict__ beta_e,
        float* __restrict__ scale_e, float* __restrict__ shift_e)
{
    const int c = threadIdx.x;          // 0..127
    float s = 0.f, q = 0.f;
    for (int wg = 0; wg < B_ * V_; ++wg) {
        s += part[wg * 256 + c];
        q += part[wg * 256 + 128 + c];
    }
    const float N    = (float)((size_t)B_ * V_ * V_);
    const float mean = s / N;
    const float var  = q / N - mean * mean;
    const float sc   = gamma_e[c] * rsqrtf(var + EPS_);
    scale_e[c] = sc;
    shift_e[c] = beta_e[c] - mean * sc;
}

// ---------------------------------------------------------------------------
// Kernel 4: e_out = relu(e_new*scale + shift) + e_in   (in place in d_out)
// ---------------------------------------------------------------------------
__global__ void __launch_bounds__(256)
efinal(const float* __restrict__ ein, float* __restrict__ eio,
       const float* __restrict__ scale_e, const float* __restrict__ shift_e)
{
    __shared__ float ssc[H_], ssh[H_];
    const int t = threadIdx.x;
    if (t < H_)        ssc[t] = scale_e[t];
    else               ssh[t - H_] = shift_e[t - H_];
    __syncthreads();

    const size_t idx = (size_t)blockIdx.x * 256 + t;             // float4 index
    const int ch4 = ((int)(idx & 31)) * 4;                       // H/4 == 32
    float4 en = ((const float4*)eio)[idx];
    float4 ei = ((const float4*)ein)[idx];
    float4 o;
    o.x = fmaxf(fmaf(en.x, ssc[ch4 + 0], ssh[ch4 + 0]), 0.f) + ei.x;
    o.y = fmaxf(fmaf(en.y, ssc[ch4 + 1], ssh[ch4 + 1]), 0.f) + ei.y;
    o.z = fmaxf(fmaf(en.z, ssc[ch4 + 2], ssh[ch4 + 2]), 0.f) + ei.z;
    o.w = fmaxf(fmaf(en.w, ssc[ch4 + 3], ssh[ch4 + 3]), 0.f) + ei.w;
    ((float4*)eio)[idx] = o;
}

// ---------------------------------------------------------------------------
// Kernel 5: h path: BN(Uh+agg) -> relu -> +h_in.  One block per channel.
// ---------------------------------------------------------------------------
__global__ void __launch_bounds__(256)
hfinal(const float* __restrict__ Uh, const float* __restrict__ agg,
       const float* __restrict__ hin,
       const float* __restrict__ gamma_h, const float* __restrict__ beta_h,
       float* __restrict__ hout)
{
    const int c = blockIdx.x;           // channel 0..127
    const int t = threadIdx.x;
    __shared__ float rsum[256], rsq[256];
    __shared__ float sc_s, sh_s;
    const int R = B_ * V_;              // 1024 rows

    float s = 0.f, q = 0.f;
    for (int r = t; r < R; r += 256) {
        const float x = Uh[r * H_ + c] + agg[r * H_ + c];
        s += x; q = fmaf(x, x, q);
    }
    rsum[t] = s; rsq[t] = q;
    __syncthreads();
    for (int ofs = 128; ofs > 0; ofs >>= 1) {
        if (t < ofs) { rsum[t] += rsum[t + ofs]; rsq[t] += rsq[t + ofs]; }
        __syncthreads();
    }
    if (t == 0) {
        const float mean = rsum[0] / (float)R;
        const float var  = rsq[0] / (float)R - mean * mean;
        const float sc   = gamma_h[c] * rsqrtf(var + EPS_);
        sc_s = sc;
        sh_s = beta_h[c] - mean * sc;
    }
    __syncthreads();
    const float sc = sc_s, sh = sh_s;
    for (int r = t; r < R; r += 256) {
        const float x = Uh[r * H_ + c] + agg[r * H_ + c];
        hout[r * H_ + c] = fmaxf(fmaf(x, sc, sh), 0.f) + hin[r * H_ + c];
    }
}

// ---------------------------------------------------------------------------
extern "C" void kernel_launch(void* const* d_in, const int* in_sizes, int n_in,
                              void* d_out, int out_size, void* d_ws, size_t ws_size,
                              hipStream_t stream) {
    const float* h   = (const float*)d_in[0];
    const float* e   = (const float*)d_in[1];
    // d_in[2] = graph (unused by the reference math)
    const float* Uw  = (const float*)d_in[3];
    const float* Ub  = (const float*)d_in[4];
    const float* Vw  = (const float*)d_in[5];
    const float* Vb  = (const float*)d_in[6];
    const float* Aw  = (const float*)d_in[7];
    const float* Ab  = (const float*)d_in[8];
    const float* Bw  = (const float*)d_in[9];
    const float* Bb  = (const float*)d_in[10];
    const float* Cw  = (const float*)d_in[11];
    const float* Cb  = (const float*)d_in[12];
    const float* gamma_h = (const float*)d_in[13];
    const float* beta_h  = (const float*)d_in[14];
    const float* gamma_e = (const float*)d_in[15];
    const float* beta_e  = (const float*)d_in[16];

    float* hout = (float*)d_out;                       // (B,V,H)
    float* eout = hout + (size_t)B_ * V_ * H_;         // (B,V,V,H), also e_new scratch

    float* ws      = (float*)d_ws;
    float* Uh      = ws;                               // 131072
    float* Vn      = Uh  + (size_t)B_ * V_ * H_;       // 131072
    float* Ahs     = Vn  + (size_t)B_ * V_ * H_;       // 131072
    float* Bhs     = Ahs + (size_t)B_ * V_ * H_;       // 131072
    float* aggw    = Bhs + (size_t)B_ * V_ * H_;       // 131072
    float* part_e  = aggw + (size_t)B_ * V_ * H_;      // 1024*256
    float* scale_e = part_e + (size_t)B_ * V_ * 256;   // 128
    float* shift_e = scale_e + 128;                    // 128

    pre_linears<<<B_ * V_, 128, 0, stream>>>(h, Uw, Ub, Vw, Vb, Aw, Ab, Bw, Bb,
                                             Uh, Vn, Ahs, Bhs);
    fused_edge<<<B_ * V_, 256, 0, stream>>>(e, Cw, Cb, Ahs, Bhs, Vn,
                                            eout, aggw, part_e);
    stats_e<<<1, 128, 0, stream>>>(part_e, gamma_e, beta_e, scale_e, shift_e);
    efinal<<<(B_ * V_ * V_ * H_) / (4 * 256), 256, 0, stream>>>(e, eout, scale_e, shift_e);
    hfinal<<<H_, 256, 0, stream>>>(Uh, aggw, h, gamma_h, beta_h, hout);
}